// LeakyTransfer_13838384628287
// MI455X (gfx1250) — compile-verified
//
#include <hip/hip_runtime.h>
#include <hip/hip_bf16.h>

// CDNA5 / gfx1250 kernel for LeakyTransfer logmatexp:
//   out = log(transfer @ exp(xx - max)) + max, transfer is 16x16
// Memory-bound (256 MB @ 23.3 TB/s ~= 11 us). The 16x16 matmul is done with
// 4 chained V_WMMA_F32_16X16X4_F32 per 16-column tile (K=16 in steps of 4).
// Transcendentals use native v_exp_f32 / v_log_f32 via amdgcn builtins (safe
// here: exp args <= 0, log args in (0,16], y >= jps > 0 since the max column
// contributes exp(0)=1 through the all-positive transfer row).

typedef __attribute__((ext_vector_type(2))) float v2f;
typedef __attribute__((ext_vector_type(8))) float v8f;

#define NSTATES 16

__global__ __launch_bounds__(256) void leaky_transfer_wmma_kernel(
    const float* __restrict__ xx,        // [16, ncols]
    const float* __restrict__ jump_rate, // [1]
    float* __restrict__ out,             // [16, ncols]
    int ncols)
{
    const int lane = threadIdx.x & 31;
    const int wave = blockIdx.x * (blockDim.x >> 5) + (threadIdx.x >> 5);
    const int c0   = wave * 16;          // first column of this wave's tile
    if (c0 >= ncols) return;             // wave-uniform exit (EXEC stays full)

    const int half = lane >> 4;          // 0: lanes 0-15, 1: lanes 16-31
    const int n    = lane & 15;          // column within tile / matrix row M

    const float jr   = jump_rate[0];
    const float jps  = jr * (1.0f / (NSTATES - 1));
    const float diag = 1.0f - jr;

    // Clamp load column on a partial tile (duplicate reads; stores guarded).
    const int col = (c0 + n < ncols) ? (c0 + n) : (ncols - 1);

    // Each lane loads 8 of its column's 16 states:
    //   k = 4*t + 2*half + j   (t = 0..3, j = 0..1)  -- the WMMA B layout.
    float x[8];
#pragma unroll
    for (int t = 0; t < 4; ++t) {
        const int k0 = 4 * t + 2 * half;
        x[2 * t + 0] = xx[(size_t)(k0 + 0) * ncols + col];
        x[2 * t + 1] = xx[(size_t)(k0 + 1) * ncols + col];
    }

    // Column max: local max of this lane's 8, combined with lane^16
    // (which holds the complementary 8 states of the same column).
    float m = x[0];
#pragma unroll
    for (int i = 1; i < 8; ++i) m = fmaxf(m, x[i]);
    m = fmaxf(m, __shfl_xor(m, 16, 32));

    // E = exp2((x - m) * log2(e)) via native v_exp_f32, kept in the
    // B-matrix register layout. Arguments are <= 0 so no range issues.
    const float LOG2E = 1.44269504088896340736f;
#pragma unroll
    for (int i = 0; i < 8; ++i)
        x[i] = __builtin_amdgcn_exp2f((x[i] - m) * LOG2E);

    // D = transfer @ E via 4 chained V_WMMA_F32_16X16X4_F32 (K = 0..15).
    // A layout: VGPR v, lane L -> A[M = L&15][K = 4t + 2*(L>>4) + v]
    // B layout: VGPR v, lane L -> B[K = 4t + 2*(L>>4) + v][N = L&15]
    v8f c = {};
#pragma unroll
    for (int t = 0; t < 4; ++t) {
        const int k0 = 4 * t + 2 * half;
        v2f a, b;
        a.x = (n == k0    ) ? diag : jps;  // A[M=n][k0]
        a.y = (n == k0 + 1) ? diag : jps;  // A[M=n][k0+1]
        b.x = x[2 * t + 0];                // B[k0][n]
        b.y = x[2 * t + 1];                // B[k0+1][n]
        c = __builtin_amdgcn_wmma_f32_16x16x4_f32(
                /*neg_a=*/false, a, /*neg_b=*/false, b,
                /*c_mod=*/(short)0, c, /*reuse_a=*/false, /*reuse_b=*/false);
    }

    // C/D layout: VGPR r, lane L -> D[M = r + 8*(L>>4)][N = L&15].
    // out[M][c0+n] = log2(D)*ln2 + m: one v_log_f32 + one v_fma_f32 each.
    const float LN2 = 0.69314718055994530942f;
    if (c0 + n < ncols) {
#pragma unroll
        for (int r = 0; r < 8; ++r) {
            const int M = r + 8 * half;
            out[(size_t)M * ncols + c0 + n] =
                fmaf(__builtin_amdgcn_logf(c[r]), LN2, m);
        }
    }
}

extern "C" void kernel_launch(void* const* d_in, const int* in_sizes, int n_in,
                              void* d_out, int out_size, void* d_ws, size_t ws_size,
                              hipStream_t stream) {
    const float* xx = (const float*)d_in[0];
    const float* jr = (const float*)d_in[1];
    float* out      = (float*)d_out;

    const int ncols = in_sizes[0] / NSTATES;          // 2,000,000
    const int cols_per_block = 16 * (256 / 32);       // 16 cols/wave * 8 waves
    const int grid = (ncols + cols_per_block - 1) / cols_per_block;

    leaky_transfer_wmma_kernel<<<grid, 256, 0, stream>>>(xx, jr, out, ncols);
}